// RouterAugmentedAttention_60413009986043
// MI455X (gfx1250) — compile-verified
//
#include <hip/hip_runtime.h>
#include <math.h>

// ---------------------------------------------------------------------------
// RouterAugmentedAttention forward, MI455X (gfx1250), fp32 end-to-end.
// All GEMMs use V_WMMA_F32_16X16X4_F32 (full fp32 on the matrix pipes).
// B=4, S=2048, D=1024, H=16, DH=64, top-K heads = 8.
// ---------------------------------------------------------------------------

#define BB 4
#define SS 2048
#define DD 1024
#define HH 16
#define DHH 64
#define TOPK 8

typedef float v2f __attribute__((ext_vector_type(2)));
typedef float v8f __attribute__((ext_vector_type(8)));

__device__ __forceinline__ v8f wmma4(v2f a, v2f b, v8f c) {
  // D = A(16x4,f32) * B(4x16,f32) + C(16x16,f32)
  return __builtin_amdgcn_wmma_f32_16x16x4_f32(
      /*neg_a=*/false, a, /*neg_b=*/false, b,
      /*c_mod=*/(short)0, c, /*reuse_a=*/false, /*reuse_b=*/false);
}

// ---------------------------------------------------------------------------
// Kernel 1: router mean -> logits -> top-8 binary head mask.  grid(B), 1024 thr
// ---------------------------------------------------------------------------
__global__ void router_kernel(const float* __restrict__ hs,
                              const float* __restrict__ rw,
                              const float* __restrict__ rb,
                              float* __restrict__ mask) {
  __shared__ float smean[DD];
  __shared__ float spart[1024];
  __shared__ float slog[HH];
  const int b = blockIdx.x;
  const int t = threadIdx.x;  // 0..1023, one column each
  float acc = 0.0f;
  const float* base = hs + (size_t)b * SS * DD + t;
  for (int s = 0; s < SS; ++s) acc += base[(size_t)s * DD];
  smean[t] = acc * (1.0f / (float)SS);
  __syncthreads();
  // logits: thread t handles head h = t&15 over 16 d-values
  const int h = t & 15, chunk = t >> 4;  // chunk 0..63
  float p = 0.0f;
  for (int j = 0; j < 16; ++j) {
    const int d = chunk + 64 * j;
    p += smean[d] * rw[d * HH + h];
  }
  spart[t] = p;
  __syncthreads();
  if (t < HH) {
    float sum = rb[t];
    for (int c = 0; c < 64; ++c) sum += spart[c * 16 + t];
    slog[t] = sum;
  }
  __syncthreads();
  if (t < HH) {
    const float v = slog[t];
    int rank = 0;
    for (int o = 0; o < HH; ++o) {
      const float ov = slog[o];
      if (ov > v || (ov == v && o < t)) rank++;  // jax top_k tie-break
    }
    mask[b * HH + t] = (rank < TOPK) ? 1.0f : 0.0f;
  }
}

// ---------------------------------------------------------------------------
// Kernel 2: QKV projection GEMM (M=8192, N=1024, K=1024) x3 via gridDim.z.
// 256 threads = 8 waves (4m x 2n), block tile 128x128, wave tile 32x64,
// K-chunk 16, fp32 WMMA.  Output layout (B,H,S,DH), scaled by head mask.
// ---------------------------------------------------------------------------
__global__ void qkv_kernel(const float* __restrict__ hs,
                           const float* __restrict__ wq, const float* __restrict__ bq,
                           const float* __restrict__ wk, const float* __restrict__ bk_,
                           const float* __restrict__ wv, const float* __restrict__ bv,
                           const float* __restrict__ mask,
                           float* __restrict__ Qo, float* __restrict__ Ko,
                           float* __restrict__ Vo) {
  __shared__ float As[128][20];  // stride 20: conflict-free A-fragment reads
  __shared__ float Bs[16][136];  // stride 136: bank = 8k+n, halves 16 apart
  const int mt = blockIdx.x * 128;
  const int nt = blockIdx.y * 128;
  const int which = blockIdx.z;
  const float* W = (which == 0) ? wq : (which == 1) ? wk : wv;
  const float* bias = (which == 0) ? bq : (which == 1) ? bk_ : bv;
  float* Out = (which == 0) ? Qo : (which == 1) ? Ko : Vo;
  const int t = threadIdx.x;
  const int wid = t >> 5, lane = t & 31;
  const int wm = wid & 3, wn = wid >> 2;  // 4 x 2 wave grid
  const int lh = lane >> 4, l16 = lane & 15;
  v8f acc[2][4] = {};
  for (int kb = 0; kb < DD; kb += 16) {
    for (int i = 0; i < 8; ++i) {
      const int e = t + i * 256;
      As[e >> 4][e & 15] = hs[(size_t)(mt + (e >> 4)) * DD + kb + (e & 15)];
    }
    for (int i = 0; i < 8; ++i) {
      const int e = t + i * 256;
      Bs[e >> 7][e & 127] = W[(size_t)(kb + (e >> 7)) * DD + nt + (e & 127)];
    }
    if (kb + 16 < DD) {  // prefetch next K-chunk (global_prefetch_b8)
      __builtin_prefetch(&hs[(size_t)(mt + (t >> 4)) * DD + kb + 16 + (t & 15)], 0, 3);
      __builtin_prefetch(&W[(size_t)(kb + 16 + (t >> 7)) * DD + nt + (t & 127)], 0, 3);
    }
    __syncthreads();
    for (int kk = 0; kk < 16; kk += 4) {
      v2f a[2], bf[4];
      for (int mi = 0; mi < 2; ++mi) {
        const int m = wm * 32 + mi * 16 + l16;
        a[mi].x = As[m][kk + 2 * lh];
        a[mi].y = As[m][kk + 1 + 2 * lh];
      }
      for (int nj = 0; nj < 4; ++nj) {
        const int n = wn * 64 + nj * 16 + l16;
        bf[nj].x = Bs[kk + 2 * lh][n];
        bf[nj].y = Bs[kk + 1 + 2 * lh][n];
      }
      for (int mi = 0; mi < 2; ++mi)
        for (int nj = 0; nj < 4; ++nj)
          acc[mi][nj] = wmma4(a[mi], bf[nj], acc[mi][nj]);
    }
    __syncthreads();
  }
  for (int mi = 0; mi < 2; ++mi)
    for (int nj = 0; nj < 4; ++nj) {
      const int ncol = nt + wn * 64 + nj * 16 + l16;
      const int hidx = ncol >> 6, dh = ncol & 63;
      const float bv_ = bias[ncol];
      for (int r = 0; r < 8; ++r) {
        const int m = mt + wm * 32 + mi * 16 + r + 8 * lh;
        const int b = m >> 11, s = m & 2047;
        const float mval = mask[b * HH + hidx];
        Out[(((size_t)(b * HH + hidx)) * SS + s) * DHH + dh] =
            (acc[mi][nj][r] + bv_) * mval;
      }
    }
}

// ---------------------------------------------------------------------------
// Kernel 3: flash attention per (b,h,qtile=64).  4 waves x 16 q-rows each.
// Skips masked heads (ctx slice = 0).  ctx layout (B,S,D) for the out GEMM.
// LDS < 64KB: P tile aliases the K tile (barrier between uses).
// ---------------------------------------------------------------------------
__global__ void attn_kernel(const float* __restrict__ Qo,
                            const float* __restrict__ Ko,
                            const float* __restrict__ Vo,
                            const float* __restrict__ am,
                            const float* __restrict__ mask,
                            float* __restrict__ ctx) {
  __shared__ float Qs[64][68];   // A-role, stride 68
  __shared__ float KPs[64][68];  // K tile (B-role transposed) then P tile (A-role)
  __shared__ float Vs[64][72];   // B-role row-reads, stride 72
  const int qt = blockIdx.x * 64;
  const int h = blockIdx.y, b = blockIdx.z;
  const int t = threadIdx.x;
  const int wid = t >> 5, lane = t & 31;
  const int lh = lane >> 4, l16 = lane & 15;
  const int rbase = wid * 16;
  const float mval = mask[b * HH + h];
  if (mval == 0.0f) {  // masked head: ctx slice is exactly zero
    for (int i = 0; i < 32; ++i) {
      const int e = t + i * 128;
      ctx[((size_t)(b * SS + qt + (e >> 6))) * DD + h * DHH + (e & 63)] = 0.0f;
    }
    return;
  }
  const float* Qb = Qo + ((size_t)(b * HH + h)) * SS * DHH;
  const float* Kb = Ko + ((size_t)(b * HH + h)) * SS * DHH;
  const float* Vb = Vo + ((size_t)(b * HH + h)) * SS * DHH;
  const float* amb = am + (size_t)b * SS;
  for (int i = 0; i < 32; ++i) {
    const int e = t + i * 128;
    Qs[e >> 6][e & 63] = Qb[(size_t)(qt + (e >> 6)) * DHH + (e & 63)];
  }
  v8f O[4] = {};
  float mrow[8], lrow[8];
  for (int r = 0; r < 8; ++r) { mrow[r] = -INFINITY; lrow[r] = 0.0f; }
  const float scale = 0.125f;  // 1/sqrt(64)
  for (int kt = 0; kt < SS; kt += 64) {
    __syncthreads();  // protect K/V (and aliased P) from previous iteration
    for (int i = 0; i < 32; ++i) {
      const int e = t + i * 128;
      const int r = e >> 6, c = e & 63;
      KPs[r][c] = Kb[(size_t)(kt + r) * DHH + c];
      Vs[r][c] = Vb[(size_t)(kt + r) * DHH + c];
    }
    if (kt + 64 < SS) {  // prefetch next K/V tile (global_prefetch_b8)
      __builtin_prefetch(&Kb[(size_t)(kt + 64 + (t >> 1)) * DHH + (t & 1) * 32], 0, 3);
      __builtin_prefetch(&Vb[(size_t)(kt + 64 + (t >> 1)) * DHH + (t & 1) * 32], 0, 3);
    }
    __syncthreads();
    // S = Q * K^T  (A = Q rows, B = K^T read lane=keypos, K-dim = dh)
    v8f sacc[4] = {};
    for (int kk = 0; kk < 64; kk += 4) {
      v2f a;
      a.x = Qs[rbase + l16][kk + 2 * lh];
      a.y = Qs[rbase + l16][kk + 1 + 2 * lh];
      for (int nj = 0; nj < 4; ++nj) {
        v2f bf;
        bf.x = KPs[nj * 16 + l16][kk + 2 * lh];
        bf.y = KPs[nj * 16 + l16][kk + 1 + 2 * lh];
        sacc[nj] = wmma4(a, bf, sacc[nj]);
      }
    }
    __syncthreads();  // everyone done reading K tile; reuse KPs for P
    float amv[4];
    for (int nj = 0; nj < 4; ++nj) amv[nj] = amb[kt + nj * 16 + l16];
    for (int r = 0; r < 8; ++r) {
      float mx = -INFINITY;
      for (int nj = 0; nj < 4; ++nj) {
        const float sv = sacc[nj][r] * scale + amv[nj];
        sacc[nj][r] = sv;
        mx = fmaxf(mx, sv);
      }
      for (int off = 1; off < 16; off <<= 1)  // reduce within 16-lane half
        mx = fmaxf(mx, __shfl_xor(mx, off, 32));
      const float mnew = fmaxf(mrow[r], mx);
      const float alpha = __expf(mrow[r] - mnew);
      float rsum = 0.0f;
      for (int nj = 0; nj < 4; ++nj) {
        const float p = __expf(sacc[nj][r] - mnew);
        sacc[nj][r] = p;
        rsum += p;
      }
      for (int off = 1; off < 16; off <<= 1)
        rsum += __shfl_xor(rsum, off, 32);
      mrow[r] = mnew;
      lrow[r] = lrow[r] * alpha + rsum;
      for (int nj = 0; nj < 4; ++nj) O[nj][r] *= alpha;
    }
    // stage P (own wave's 16 rows) into aliased buffer
    for (int nj = 0; nj < 4; ++nj)
      for (int r = 0; r < 8; ++r)
        KPs[rbase + r + 8 * lh][nj * 16 + l16] = sacc[nj][r];
    // O += P * V  (A = P rows, B = V rows, K-dim = keypos)
    for (int kk = 0; kk < 64; kk += 4) {
      v2f a;
      a.x = KPs[rbase + l16][kk + 2 * lh];
      a.y = KPs[rbase + l16][kk + 1 + 2 * lh];
      for (int nj = 0; nj < 4; ++nj) {
        v2f bf;
        bf.x = Vs[kk + 2 * lh][nj * 16 + l16];
        bf.y = Vs[kk + 1 + 2 * lh][nj * 16 + l16];
        O[nj] = wmma4(a, bf, O[nj]);
      }
    }
  }
  for (int nj = 0; nj < 4; ++nj) {
    const int dh = nj * 16 + l16;
    for (int r = 0; r < 8; ++r) {
      const int srow = qt + rbase + r + 8 * lh;
      ctx[((size_t)(b * SS + srow)) * DD + h * DHH + dh] = O[nj][r] / lrow[r];
    }
  }
}

// ---------------------------------------------------------------------------
// Kernel 4: out = ctx @ wo + bo + hidden  (pre-LN, written to d_out)
// 256 threads, block tile 128x128, same scheme as kernel 2.
// ---------------------------------------------------------------------------
__global__ void oproj_kernel(const float* __restrict__ ctxp,
                             const float* __restrict__ wo,
                             const float* __restrict__ bo,
                             const float* __restrict__ hs,
                             float* __restrict__ out) {
  __shared__ float As[128][20];
  __shared__ float Bs[16][136];
  const int mt = blockIdx.x * 128;
  const int nt = blockIdx.y * 128;
  const int t = threadIdx.x;
  const int wid = t >> 5, lane = t & 31;
  const int wm = wid & 3, wn = wid >> 2;
  const int lh = lane >> 4, l16 = lane & 15;
  v8f acc[2][4] = {};
  for (int kb = 0; kb < DD; kb += 16) {
    for (int i = 0; i < 8; ++i) {
      const int e = t + i * 256;
      As[e >> 4][e & 15] = ctxp[(size_t)(mt + (e >> 4)) * DD + kb + (e & 15)];
    }
    for (int i = 0; i < 8; ++i) {
      const int e = t + i * 256;
      Bs[e >> 7][e & 127] = wo[(size_t)(kb + (e >> 7)) * DD + nt + (e & 127)];
    }
    if (kb + 16 < DD) {
      __builtin_prefetch(&ctxp[(size_t)(mt + (t >> 4)) * DD + kb + 16 + (t & 15)], 0, 3);
      __builtin_prefetch(&wo[(size_t)(kb + 16 + (t >> 7)) * DD + nt + (t & 127)], 0, 3);
    }
    __syncthreads();
    for (int kk = 0; kk < 16; kk += 4) {
      v2f a[2], bf[4];
      for (int mi = 0; mi < 2; ++mi) {
        const int m = wm * 32 + mi * 16 + l16;
        a[mi].x = As[m][kk + 2 * lh];
        a[mi].y = As[m][kk + 1 + 2 * lh];
      }
      for (int nj = 0; nj < 4; ++nj) {
        const int n = wn * 64 + nj * 16 + l16;
        bf[nj].x = Bs[kk + 2 * lh][n];
        bf[nj].y = Bs[kk + 1 + 2 * lh][n];
      }
      for (int mi = 0; mi < 2; ++mi)
        for (int nj = 0; nj < 4; ++nj)
          acc[mi][nj] = wmma4(a[mi], bf[nj], acc[mi][nj]);
    }
    __syncthreads();
  }
  for (int mi = 0; mi < 2; ++mi)
    for (int nj = 0; nj < 4; ++nj) {
      const int n = nt + wn * 64 + nj * 16 + l16;
      const float bias_v = bo[n];
      for (int r = 0; r < 8; ++r) {
        const int m = mt + wm * 32 + mi * 16 + r + 8 * lh;
        out[(size_t)m * DD + n] =
            acc[mi][nj][r] + bias_v + hs[(size_t)m * DD + n];
      }
    }
}

// ---------------------------------------------------------------------------
// Kernel 5: in-place LayerNorm per row (two-pass for non-negative variance)
// ---------------------------------------------------------------------------
__global__ void ln_kernel(float* __restrict__ out,
                          const float* __restrict__ g,
                          const float* __restrict__ be) {
  __shared__ float sbuf[8];
  const int t = threadIdx.x;  // 256 threads
  float* rp = out + (size_t)blockIdx.x * DD;
  float x[4];
  for (int i = 0; i < 4; ++i) x[i] = rp[t + i * 256];
  float s = x[0] + x[1] + x[2] + x[3];
  for (int off = 1; off < 32; off <<= 1) s += __shfl_xor(s, off, 32);
  if ((t & 31) == 0) sbuf[t >> 5] = s;
  __syncthreads();
  float tot = 0.0f;
  for (int i = 0; i < 8; ++i) tot += sbuf[i];
  const float mu = tot * (1.0f / (float)DD);
  __syncthreads();
  float vs = 0.0f;
  for (int i = 0; i < 4; ++i) { const float d = x[i] - mu; vs += d * d; }
  for (int off = 1; off < 32; off <<= 1) vs += __shfl_xor(vs, off, 32);
  if ((t & 31) == 0) sbuf[t >> 5] = vs;
  __syncthreads();
  float vt = 0.0f;
  for (int i = 0; i < 8; ++i) vt += sbuf[i];
  const float var = vt * (1.0f / (float)DD);
  const float inv = 1.0f / sqrtf(var + 1e-12f);
  for (int i = 0; i < 4; ++i) {
    const int c = t + i * 256;
    rp[c] = (x[i] - mu) * inv * g[c] + be[c];
  }
}

// ---------------------------------------------------------------------------
extern "C" void kernel_launch(void* const* d_in, const int* in_sizes, int n_in,
                              void* d_out, int out_size, void* d_ws, size_t ws_size,
                              hipStream_t stream) {
  const float* hs = (const float*)d_in[0];
  const float* am = (const float*)d_in[1];
  const float* rw = (const float*)d_in[2];
  const float* rb = (const float*)d_in[3];
  const float* wq = (const float*)d_in[4];
  const float* bq = (const float*)d_in[5];
  const float* wk = (const float*)d_in[6];
  const float* bk = (const float*)d_in[7];
  const float* wv = (const float*)d_in[8];
  const float* bv = (const float*)d_in[9];
  const float* wo = (const float*)d_in[10];
  const float* bo = (const float*)d_in[11];
  const float* lg = (const float*)d_in[12];
  const float* lb = (const float*)d_in[13];
  float* ws = (float*)d_ws;
  const size_t QSZ = (size_t)BB * HH * SS * DHH;  // 8,388,608 floats
  float* mask = ws;            // 64 floats (256-float slot for alignment)
  float* Q = ws + 256;
  float* K = Q + QSZ;
  float* V = K + QSZ;
  float* ctx = V + QSZ;        // total ~128 MiB workspace
  float* out = (float*)d_out;

  router_kernel<<<BB, 1024, 0, stream>>>(hs, rw, rb, mask);
  qkv_kernel<<<dim3((BB * SS) / 128, DD / 128, 3), 256, 0, stream>>>(
      hs, wq, bq, wk, bk, wv, bv, mask, Q, K, V);
  attn_kernel<<<dim3(SS / 64, HH, BB), 128, 0, stream>>>(Q, K, V, am, mask, ctx);
  oproj_kernel<<<dim3((BB * SS) / 128, DD / 128), 256, 0, stream>>>(ctx, wo, bo, hs, out);
  ln_kernel<<<BB * SS, 256, 0, stream>>>(out, lg, lb);
}